// ComplexFaberConv_57174604644564
// MI455X (gfx1250) — compile-verified
//
#include <hip/hip_runtime.h>
#include <math.h>

// CDNA5 WMMA fp32: D(16x16) = A(16x4) x B(4x16) + C
typedef __attribute__((ext_vector_type(2))) float v2f;
typedef __attribute__((ext_vector_type(8))) float v8f;

#define D_FEAT 64

// ---------------- zero init (ws accumulators + d_out accumulators) ----------
__global__ void zero2_kernel(float* __restrict__ a, long na4,
                             float* __restrict__ b, long nb4) {
    long i = (long)blockIdx.x * blockDim.x + threadIdx.x;
    long stride = (long)gridDim.x * blockDim.x;
    float4 z = make_float4(0.f, 0.f, 0.f, 0.f);
    for (long j = i; j < na4 + nb4; j += stride) {
        if (j < na4) ((float4*)a)[j] = z;
        else         ((float4*)b)[j - na4] = z;
    }
}

// ---------------- degree scatter -------------------------------------------
__global__ void degree_kernel(const int* __restrict__ row, const int* __restrict__ col,
                              int E, float* degO, float* degI) {
    int e = blockIdx.x * blockDim.x + threadIdx.x;
    if (e >= E) return;
    unsafeAtomicAdd(&degO[row[e]], 1.0f);
    unsafeAtomicAdd(&degI[col[e]], 1.0f);
}

// ---------------- deg -> deg^-0.25 in place --------------------------------
__global__ void deginv_kernel(float* degO, float* degI, int n) {
    int i = blockIdx.x * blockDim.x + threadIdx.x;
    if (i >= n) return;
    float d0 = degO[i];
    float d1 = degI[i];
    degO[i] = (d0 > 0.f) ? powf(d0, -0.25f) : 0.f;
    degI[i] = (d1 > 0.f) ? powf(d1, -0.25f) : 0.f;
}

// ---------------- fold K orders + alpha into combined weights/biases --------
__global__ void wcomb_kernel(const float* __restrict__ Wr, const float* __restrict__ br,
                             const float* __restrict__ Wi, const float* __restrict__ bi,
                             int K, float* WRc, float* WIc, float* bmi, float* bpi) {
    int i = blockIdx.x * blockDim.x + threadIdx.x;
    if (i < D_FEAT * D_FEAT) {
        float sr = 0.f, si = 0.f, c = 1.f;
        for (int k = 0; k < K; ++k) {
            sr += c * Wr[k * D_FEAT * D_FEAT + i];
            si += c * Wi[k * D_FEAT * D_FEAT + i];
            c *= 0.5f;
        }
        WRc[i] = sr;
        WIc[i] = si;
    } else if (i < D_FEAT * D_FEAT + D_FEAT) {
        int o = i - D_FEAT * D_FEAT;
        float sr = 0.f, si = 0.f, c = 1.f;
        for (int k = 0; k < K; ++k) {
            sr += c * br[k * D_FEAT + o];
            si += c * bi[k * D_FEAT + o];
            c *= 0.5f;
        }
        bmi[o] = sr - si;   // bR - bI
        bpi[o] = sr + si;   // bR + bI
    }
}

// ---------------- edge SpMM scatter: ONE WAVE per edge ----------------------
// Edge id is wave-uniform (readfirstlane) so row/col/degree loads become
// scalar (s_load, KMcnt) instead of 32 redundant vector loads. Each lane
// covers 2 dims via float2: 4x b64 coalesced gathers + 8 L2 f32 atomics.
__global__ __launch_bounds__(256) void spmm_kernel(
        const float* __restrict__ xr, const float* __restrict__ xi,
        const int* __restrict__ row, const int* __restrict__ col, int E,
        const float* __restrict__ out_inv, const float* __restrict__ in_inv,
        float* yr, float* yi, float* yrt, float* yit) {
    const int lane = threadIdx.x & 31;
    int e = blockIdx.x * 8 + (threadIdx.x >> 5);
    e = __builtin_amdgcn_readfirstlane(e);        // force SGPR -> scalar loads
    if (e >= E) return;

    const int r = row[e];
    const int c = col[e];
    const float w = out_inv[r] * in_inv[c];

    const int d = lane << 1;
    const long ro = (long)r * D_FEAT + d;
    const long co = (long)c * D_FEAT + d;

    const float2 xrc = *(const float2*)(xr + co);
    const float2 xic = *(const float2*)(xi + co);
    const float2 xrr = *(const float2*)(xr + ro);
    const float2 xir = *(const float2*)(xi + ro);

    unsafeAtomicAdd(&yr[ro],      w * xrc.x);   // adj_norm   @ x_real
    unsafeAtomicAdd(&yr[ro + 1],  w * xrc.y);
    unsafeAtomicAdd(&yi[ro],      w * xic.x);   // adj_norm   @ x_imag
    unsafeAtomicAdd(&yi[ro + 1],  w * xic.y);
    unsafeAtomicAdd(&yrt[co],     w * xrr.x);   // adj_t_norm @ x_real
    unsafeAtomicAdd(&yrt[co + 1], w * xrr.y);
    unsafeAtomicAdd(&yit[co],     w * xir.x);   // adj_t_norm @ x_imag
    unsafeAtomicAdd(&yit[co + 1], w * xir.y);
}

// ---------------- fused WMMA GEMM + complex combine (in place over d_out) ---
// Block = 128 threads (4 waves); block b owns nodes [16b, 16b+16).
// Wave w computes output dims [16w, 16w+16) for real and imag via 4 chains of
// 16x V_WMMA_F32_16X16X4_F32: ZR@WR^T, ZI@WI^T, y_real@WI^T, ZI@WR^T.
__global__ __launch_bounds__(128) void faber_gemm_kernel(
        const float* __restrict__ yr, const float* __restrict__ yi,
        const float* __restrict__ WRc, const float* __restrict__ WIc,
        const float* __restrict__ bmi, const float* __restrict__ bpi,
        float* outR, float* outI) {
    const int wave = threadIdx.x >> 5;
    const int lane = threadIdx.x & 31;
    const int l16  = lane & 15;
    const int half = lane >> 4;           // 0: K=k..k+1, 1: K=k+2..k+3 (A/B frag split)

    const float* yrt = outR;              // transpose accumulators live in d_out
    const float* yit = outI;

    const long rowOff = (long)(blockIdx.x * 16 + l16) * D_FEAT;  // A row (node)
    const long wOff   = (long)(wave * 16 + l16) * D_FEAT;        // B row (out dim)

    v8f accRa = {};  // ZR @ WR^T
    v8f accRb = {};  // ZI @ WI^T
    v8f accIa = {};  // y_real @ WI^T
    v8f accIb = {};  // ZI @ WR^T

    #pragma unroll
    for (int s = 0; s < 16; ++s) {
        const int k = 4 * s + 2 * half;   // this lane's two K values (branch-free)
        float2 ar  = *(const float2*)(yr  + rowOff + k);
        float2 art = *(const float2*)(yrt + rowOff + k);
        float2 ai  = *(const float2*)(yi  + rowOff + k);
        float2 ait = *(const float2*)(yit + rowOff + k);
        v2f Azr = {0.5f * (ar.x + art.x), 0.5f * (ar.y + art.y)};
        v2f Azi = {0.5f * (ai.x + ait.x), 0.5f * (ai.y + ait.y)};
        v2f Ayr = {ar.x, ar.y};
        float2 wr2 = *(const float2*)(WRc + wOff + k);
        float2 wi2 = *(const float2*)(WIc + wOff + k);
        v2f Bwr = {wr2.x, wr2.y};
        v2f Bwi = {wi2.x, wi2.y};
        accRa = __builtin_amdgcn_wmma_f32_16x16x4_f32(false, Azr, false, Bwr, (short)0, accRa, false, false);
        accRb = __builtin_amdgcn_wmma_f32_16x16x4_f32(false, Azi, false, Bwi, (short)0, accRb, false, false);
        accIa = __builtin_amdgcn_wmma_f32_16x16x4_f32(false, Ayr, false, Bwi, (short)0, accIa, false, false);
        accIb = __builtin_amdgcn_wmma_f32_16x16x4_f32(false, Azi, false, Bwr, (short)0, accIb, false, false);
    }

    const int colD = wave * 16 + l16;     // output dim (N of the D tile)
    const float bm = bmi[colD];
    const float bp = bpi[colD];

    // all waves finished READING d_out-resident yrt/yit for this node tile
    __syncthreads();

    #pragma unroll
    for (int r = 0; r < 8; ++r) {
        long nodeM = (long)(blockIdx.x * 16 + r + 8 * half);   // D row (M) layout
        outR[nodeM * D_FEAT + colD] = accRa[r] - accRb[r] + bm;
        outI[nodeM * D_FEAT + colD] = accIa[r] + accIb[r] + bp;
    }
}

extern "C" void kernel_launch(void* const* d_in, const int* in_sizes, int n_in,
                              void* d_out, int out_size, void* d_ws, size_t ws_size,
                              hipStream_t stream) {
    const float* xr = (const float*)d_in[0];
    const float* xi = (const float*)d_in[1];
    const int*   ei = (const int*)d_in[2];
    const float* Wr = (const float*)d_in[3];
    const float* br = (const float*)d_in[4];
    const float* Wi = (const float*)d_in[5];
    const float* bi = (const float*)d_in[6];

    const int n = in_sizes[0] / D_FEAT;        // 100000
    const int E = in_sizes[2] / 2;             // 1600000
    const int K = in_sizes[3] / (D_FEAT * D_FEAT);  // 3
    const long NF = (long)n * D_FEAT;

    // workspace layout (floats): yr | yi | degO | degI | WRc | WIc | bmi | bpi
    float* ws   = (float*)d_ws;
    float* yr   = ws;
    float* yi_  = ws + NF;
    float* degO = ws + 2 * NF;
    float* degI = degO + n;
    float* WRc  = degI + n;
    float* WIc  = WRc + D_FEAT * D_FEAT;
    float* bmi  = WIc + D_FEAT * D_FEAT;
    float* bpi  = bmi + D_FEAT;

    float* outR = (float*)d_out;               // doubles as y_real_t accumulator
    float* outI = outR + NF;                   // doubles as y_imag_t accumulator
    const int* row = ei;
    const int* col = ei + E;

    // 1) zero accumulators (yr, yi, degrees in ws; yrt, yit in d_out)
    zero2_kernel<<<2048, 256, 0, stream>>>(ws, (2 * NF + 2 * (long)n) / 4,
                                           outR, (2 * NF) / 4);
    // 2) degrees
    degree_kernel<<<(E + 255) / 256, 256, 0, stream>>>(row, col, E, degO, degI);
    // 3) deg^-0.25
    deginv_kernel<<<(n + 255) / 256, 256, 0, stream>>>(degO, degI, n);
    // 4) fold weights/biases across K orders
    wcomb_kernel<<<(D_FEAT * D_FEAT + D_FEAT + 255) / 256, 256, 0, stream>>>(
        Wr, br, Wi, bi, K, WRc, WIc, bmi, bpi);
    // 5) edge scatter (4 SpMMs fused, one wave per edge, scalar index loads)
    spmm_kernel<<<(E + 7) / 8, 256, 0, stream>>>(xr, xi, row, col, E,
                                                 degO, degI, yr, yi_, outR, outI);
    // 6) fused fp32 WMMA GEMM + complex/alpha combine, in place over d_out
    faber_gemm_kernel<<<n / 16, 128, 0, stream>>>(yr, yi_, WRc, WIc, bmi, bpi,
                                                  outR, outI);
}